// RotorQuantCompressor_18605798327027
// MI455X (gfx1250) — compile-verified
//
#include <hip/hip_runtime.h>
#include <stdint.h>

// Problem constants (from reference): B=8, S=2048, DIM_HEAD=128, BITS=3
#define DIM_HEAD       128
#define N_GROUPS       43          // (128+2)/3
#define ROW_ELEMS      129         // padded row = N_GROUPS*3
#define LDS_STRIDE     132         // floats; 528 B rows -> 16B aligned for b128 DMA
#define CHUNKS_PER_ROW 32          // 128 floats / 4 per 16B chunk
#define BLOCK          64          // 2 wave32s
#define ROWS_PER_BLOCK 64          // one row per thread

// ---- CDNA5 async global<->LDS helpers (ASYNCcnt path) ----
__device__ __forceinline__ uint32_t lds_off(const void* p) {
    // flat LDS pointer = {aperture_hi, lds_byte_offset}; low 32 bits are the DS offset
    return (uint32_t)(uintptr_t)p;
}

__device__ __forceinline__ void async_load_b128(uint32_t lds_byte, const float* gsrc) {
    uint64_t ga = (uint64_t)(uintptr_t)gsrc;
    asm volatile("global_load_async_to_lds_b128 %0, %1, off"
                 :: "v"(lds_byte), "v"(ga)
                 : "memory");
}

__device__ __forceinline__ void async_store_b128(float* gdst, uint32_t lds_byte) {
    uint64_t ga = (uint64_t)(uintptr_t)gdst;
    asm volatile("global_store_async_from_lds_b128 %0, %1, off"
                 :: "v"(ga), "v"(lds_byte)
                 : "memory");
}

__device__ __forceinline__ void wait_async0() {
    asm volatile("s_wait_asynccnt 0" ::: "memory");
}

__global__ __launch_bounds__(BLOCK)
void rotor_quant_roundtrip_kernel(const float* __restrict__ x,
                                  const float* __restrict__ M,
                                  const float* __restrict__ centroids,
                                  float* __restrict__ out,
                                  int n_rows)
{
    __shared__ float tile[ROWS_PER_BLOCK * LDS_STRIDE];
    __shared__ float Msh[N_GROUPS * 9];

    const int tid      = threadIdx.x;
    const int rowBase  = blockIdx.x * ROWS_PER_BLOCK;
    const int rowsHere = min(ROWS_PER_BLOCK, n_rows - rowBase);
    const int totalChunks = rowsHere * CHUNKS_PER_ROW;   // 16B chunks in the tile

    // Stage the 43 rotation matrices (1.5 KB) into LDS.
    for (int i = tid; i < N_GROUPS * 9; i += BLOCK) Msh[i] = M[i];

    // Async DMA in: 64 rows x 128 floats as 512B/wave b128 transfers.
    // 32 loads issued per wave -> 16KB in flight per wave.
    const float* src = x + (size_t)rowBase * DIM_HEAD;
    for (int k = tid; k < totalChunks; k += BLOCK) {
        int r  = k >> 5;        // k / 32 : local row
        int c4 = k & 31;        // 16B chunk within row
        async_load_b128(lds_off(&tile[r * LDS_STRIDE + c4 * 4]), src + k * 4);
    }
    // Zero the pad slot (element 128) of this thread's row (disjoint from async writes).
    if (tid < rowsHere) tile[tid * LDS_STRIDE + DIM_HEAD] = 0.0f;

    // Uniform quantizer parameters (centroids are linspace: exact uniform spacing).
    const float c0        = centroids[0];
    const float delta     = (centroids[7] - centroids[0]) * (1.0f / 7.0f);
    const float inv_delta = 1.0f / delta;

    wait_async0();        // my wave's async loads landed in LDS
    __syncthreads();      // all waves' loads + pad zeroing visible

    if (tid < rowsHere) {
        float* row = &tile[tid * LDS_STRIDE];

        // Pass 1: squared norm over the 128 real elements.
        float ss = 0.0f;
        #pragma unroll 4
        for (int i = 0; i < DIM_HEAD; ++i) {
            float v = row[i];
            ss = fmaf(v, v, ss);
        }
        float nrm = fmaxf(sqrtf(ss), 1e-8f);
        float inv = 1.0f / nrm;

        // Pass 2: per-group rotate -> quantize/dequantize -> derotate -> rescale, in place.
        for (int g = 0; g < N_GROUPS; ++g) {
            const float* m = &Msh[g * 9];   // m[3*i + j] = M[g][i][j]
            float m00 = m[0], m01 = m[1], m02 = m[2];
            float m10 = m[3], m11 = m[4], m12 = m[5];
            float m20 = m[6], m21 = m[7], m22 = m[8];

            float u0 = row[3 * g + 0] * inv;
            float u1 = row[3 * g + 1] * inv;
            float u2 = row[3 * g + 2] * inv;

            // rotated_j = sum_i u_i * M[i][j]
            float r0 = fmaf(u0, m00, fmaf(u1, m10, u2 * m20));
            float r1 = fmaf(u0, m01, fmaf(u1, m11, u2 * m21));
            float r2 = fmaf(u0, m02, fmaf(u1, m12, u2 * m22));

            // uniform quantize/dequantize: deq = c0 + clamp(round((v-c0)/delta),0,7)*delta
            float k0 = fminf(fmaxf(rintf((r0 - c0) * inv_delta), 0.0f), 7.0f);
            float k1 = fminf(fmaxf(rintf((r1 - c0) * inv_delta), 0.0f), 7.0f);
            float k2 = fminf(fmaxf(rintf((r2 - c0) * inv_delta), 0.0f), 7.0f);
            float q0 = fmaf(k0, delta, c0);
            float q1 = fmaf(k1, delta, c0);
            float q2 = fmaf(k2, delta, c0);

            // derot_j = sum_i q_i * M[j][i]
            float d0 = fmaf(q0, m00, fmaf(q1, m01, q2 * m02));
            float d1 = fmaf(q0, m10, fmaf(q1, m11, q2 * m12));
            float d2 = fmaf(q0, m20, fmaf(q1, m21, q2 * m22));

            row[3 * g + 0] = d0 * nrm;
            row[3 * g + 1] = d1 * nrm;
            row[3 * g + 2] = d2 * nrm;
        }
    }

    __syncthreads();      // compute results visible to the cooperative store phase

    // Async DMA out: 512B/wave b128 stores straight from LDS (pad cols 128..131 skipped).
    float* dst = out + (size_t)rowBase * DIM_HEAD;
    for (int k = tid; k < totalChunks; k += BLOCK) {
        int r  = k >> 5;
        int c4 = k & 31;
        async_store_b128(dst + k * 4, lds_off(&tile[r * LDS_STRIDE + c4 * 4]));
    }
    // S_ENDPGM performs an implicit wait-idle, draining ASYNCcnt before LDS teardown.
}

extern "C" void kernel_launch(void* const* d_in, const int* in_sizes, int n_in,
                              void* d_out, int out_size, void* d_ws, size_t ws_size,
                              hipStream_t stream) {
    const float* x         = (const float*)d_in[0];   // (B, S, 128) f32
    const float* M         = (const float*)d_in[1];   // (43, 3, 3)  f32
    const float* centroids = (const float*)d_in[2];   // (8,)        f32
    float* out             = (float*)d_out;           // (B, S, 128) f32

    int n_rows   = in_sizes[0] / DIM_HEAD;                          // 16384
    int n_blocks = (n_rows + ROWS_PER_BLOCK - 1) / ROWS_PER_BLOCK;  // 256

    rotor_quant_roundtrip_kernel<<<n_blocks, BLOCK, 0, stream>>>(x, M, centroids, out, n_rows);
}